// AttentionBlock_1906965479948
// MI455X (gfx1250) — compile-verified
//
#include <hip/hip_runtime.h>
#include <math.h>

#define GROUPS 32
#define HEADS  8
#define CDIM   512
#define NSPAT  1024
#define HDIM   32

typedef __attribute__((ext_vector_type(2))) float v2f;
typedef __attribute__((ext_vector_type(8))) float v8f;

// D = A(16x4) * B(4x16) + C, full fp32 (V_WMMA_F32_16X16X4_F32)
__device__ __forceinline__ v8f wmma4(v2f a, v2f b, v8f c) {
  return __builtin_amdgcn_wmma_f32_16x16x4_f32(false, a, false, b, (short)0, c,
                                               false, false);
}

__device__ __forceinline__ float redmax16(float v) {
  v = fmaxf(v, __shfl_xor(v, 1, 32));
  v = fmaxf(v, __shfl_xor(v, 2, 32));
  v = fmaxf(v, __shfl_xor(v, 4, 32));
  v = fmaxf(v, __shfl_xor(v, 8, 32));
  return v;
}
__device__ __forceinline__ float redsum16(float v) {
  v += __shfl_xor(v, 1, 32);
  v += __shfl_xor(v, 2, 32);
  v += __shfl_xor(v, 4, 32);
  v += __shfl_xor(v, 8, 32);
  return v;
}

// ---------------------------------------------------------------------------
// Simple transpose: in [R, C] -> outT [C, R]  (one-time weight prep)
// ---------------------------------------------------------------------------
__global__ void transpose_kernel(const float* __restrict__ in,
                                 float* __restrict__ outT, int R, int Ccols) {
  int t = blockIdx.x * 256 + threadIdx.x;
  if (t >= R * Ccols) return;
  int r = t / Ccols, c = t % Ccols;
  outT[(size_t)c * R + r] = in[t];
}

// ---------------------------------------------------------------------------
// 0.1 * cos(2*pi*dist/(|wl|*H + 1e-6)) table, [NSPAT, NSPAT]
// ---------------------------------------------------------------------------
__global__ void interf_kernel(const float* __restrict__ wl_ptr,
                              float* __restrict__ interf) {
  int t = blockIdx.x * blockDim.x + threadIdx.x;
  if (t >= NSPAT * NSPAT) return;
  int i = t >> 10, j = t & 1023;
  float dy = (float)((i >> 5) - (j >> 5));
  float dx = (float)((i & 31) - (j & 31));
  float dist = sqrtf(dy * dy + dx * dx + 1e-8f);
  float denom = fabsf(wl_ptr[0]) * (float)HDIM + 1e-6f;
  interf[t] = 0.1f * cosf(6.283185307179586f * dist / denom);
}

// ---------------------------------------------------------------------------
// GroupNorm: in (spatial_major? [B,N,C] : [B,C,N]) -> out [B,N,C]
// one block per (b, group); 16 channels x 1024 spatial
// ---------------------------------------------------------------------------
__global__ void groupnorm_kernel(const float* __restrict__ x,
                                 const float* __restrict__ gw,
                                 const float* __restrict__ gb,
                                 float* __restrict__ out, int spatial_major) {
  __shared__ float s1[256], s2[256];
  int b = blockIdx.x >> 5;
  int g = blockIdx.x & 31;
  int tid = threadIdx.x;
  float sum = 0.f, sq = 0.f;
  for (int f = tid; f < 16 * NSPAT; f += 256) {
    int c = f >> 10, n = f & 1023;
    int ch = g * 16 + c;
    size_t idx = spatial_major ? ((size_t)(b * NSPAT + n) * CDIM + ch)
                               : ((size_t)(b * CDIM + ch) * NSPAT + n);
    float v = x[idx];
    sum += v;
    sq += v * v;
  }
  s1[tid] = sum;
  s2[tid] = sq;
  __syncthreads();
  for (int off = 128; off > 0; off >>= 1) {
    if (tid < off) {
      s1[tid] += s1[tid + off];
      s2[tid] += s2[tid + off];
    }
    __syncthreads();
  }
  float mu = s1[0] * (1.0f / 16384.0f);
  float var = s2[0] * (1.0f / 16384.0f) - mu * mu;
  float rs = rsqrtf(var + 1e-5f);
  for (int f = tid; f < 16 * NSPAT; f += 256) {
    int c = f >> 10, n = f & 1023;
    int ch = g * 16 + c;
    size_t idx = spatial_major ? ((size_t)(b * NSPAT + n) * CDIM + ch)
                               : ((size_t)(b * CDIM + ch) * NSPAT + n);
    out[(size_t)(b * NSPAT + n) * CDIM + ch] =
        (x[idx] - mu) * rs * gw[ch] + gb[ch];
  }
}

// ---------------------------------------------------------------------------
// Batched GEMM via V_WMMA_F32_16X16X4_F32, weights pre-TRANSPOSED.
//   A:  [B, M, K] (row stride lda, batch stride a_bs)
//   WT: [Nc, K]   (row stride ldwt == K) -> B-fragments are contiguous b64
// Each wave computes a 16x64 tile (4 accumulators reuse one A frag).
// Inner loop is software-pipelined (next chunk's 5 b64 loads are in flight
// while the current chunk's 4 WMMAs execute).
// mode 0: out[b,m,col]          (row stride ldo)
// mode 1: out[b,m,col] = acc + bias[col] + resid[b,col,m]   (resid [B,Nc,M])
// mode 2: out[b,col,m] = acc + bias[col] + resid[b,m,col]   (transposed store)
// mode 3: out[b,col,m] = acc                                 (transposed store)
// ---------------------------------------------------------------------------
__global__ void gemm_wmma_kernel(const float* __restrict__ A, int lda, long a_bs,
                                 const float* __restrict__ WT, int ldwt,
                                 const float* __restrict__ bias,
                                 const float* __restrict__ resid,
                                 float* __restrict__ out, int ldo, int M, int K,
                                 int Nc, int mode) {
  int lane = threadIdx.x & 31;
  int half = lane >> 4, l15 = lane & 15;
  int wid = threadIdx.x >> 5;
  int tilesM = (M + 15) >> 4;
  int tilesN = Nc >> 6;  // Nc is a multiple of 64 in all calls
  int t = blockIdx.x * 8 + wid;
  if (t >= tilesM * tilesN) return;  // wave-uniform
  int tm = t % tilesM, tn = t / tilesM;
  int b = blockIdx.z;
  const float* Ab = A + (size_t)b * a_bs;
  int arow = tm * 16 + l15;
  int arow_c = arow < M ? arow : (M - 1);  // rows are independent in WMMA
  int colb = tn * 64;
  const float* ar = Ab + (size_t)arow_c * lda + 2 * half;
  const float* wt = WT + (size_t)(colb + l15) * ldwt + 2 * half;
  size_t wstep = (size_t)16 * ldwt;
  v8f acc0 = {0, 0, 0, 0, 0, 0, 0, 0};
  v8f acc1 = acc0, acc2 = acc0, acc3 = acc0;
  // software-pipelined main loop
  v2f a_c  = *(const v2f*)(ar);
  v2f b0_c = *(const v2f*)(wt);
  v2f b1_c = *(const v2f*)(wt + wstep);
  v2f b2_c = *(const v2f*)(wt + 2 * wstep);
  v2f b3_c = *(const v2f*)(wt + 3 * wstep);
#pragma unroll 4
  for (int k0 = 4; k0 < K; k0 += 4) {
    v2f a_n  = *(const v2f*)(ar + k0);
    v2f b0_n = *(const v2f*)(wt + k0);
    v2f b1_n = *(const v2f*)(wt + wstep + k0);
    v2f b2_n = *(const v2f*)(wt + 2 * wstep + k0);
    v2f b3_n = *(const v2f*)(wt + 3 * wstep + k0);
    acc0 = wmma4(a_c, b0_c, acc0);
    acc1 = wmma4(a_c, b1_c, acc1);
    acc2 = wmma4(a_c, b2_c, acc2);
    acc3 = wmma4(a_c, b3_c, acc3);
    a_c = a_n;
    b0_c = b0_n;
    b1_c = b1_n;
    b2_c = b2_n;
    b3_c = b3_n;
  }
  acc0 = wmma4(a_c, b0_c, acc0);
  acc1 = wmma4(a_c, b1_c, acc1);
  acc2 = wmma4(a_c, b2_c, acc2);
  acc3 = wmma4(a_c, b3_c, acc3);

  v8f accs[4] = {acc0, acc1, acc2, acc3};
#pragma unroll
  for (int j = 0; j < 4; j++) {
#pragma unroll
    for (int r = 0; r < 8; r++) {
      int m = tm * 16 + r + 8 * half;
      if (m >= M) continue;
      int col = colb + j * 16 + l15;
      float val = accs[j][r];
      if (mode == 0) {
        out[((size_t)b * M + m) * ldo + col] = val;
      } else if (mode == 1) {
        val += bias[col] + resid[((size_t)b * Nc + col) * M + m];
        out[((size_t)b * M + m) * ldo + col] = val;
      } else if (mode == 2) {
        val += bias[col] + resid[((size_t)b * M + m) * Nc + col];
        out[((size_t)b * Nc + col) * ldo + m] = val;
      } else {  // mode 3
        out[((size_t)b * Nc + col) * ldo + m] = val;
      }
    }
  }
}

// ---------------------------------------------------------------------------
// Flash attention, per-wave 16-query tile, dh = 64 fixed, 8 heads.
// Q/K element (b, n, d) at base + b*bs + h*64 + n*ld + d.
// V is TRANSPOSED: element (b, d, n) at VT + b*vt_bs + (h*64 + d)*vt_ld + n,
// so both K^T and V B-fragments are contiguous b64 loads per lane.
// interf != nullptr adds interf[i*NSPAT + j] to the (pre-scaled) logits.
// Online softmax with 16-lane shfl reductions; P re-layout via per-wave LDS.
// Out-of-range keys: S columns forced to -1e30 -> P columns exactly 0, so
// padded/garbage (finite) V values contribute exactly 0.
// ---------------------------------------------------------------------------
__global__ void flash_wmma_kernel(const float* __restrict__ Q, long q_bs, int q_ld,
                                  const float* __restrict__ Kp, long k_bs, int k_ld,
                                  const float* __restrict__ VT, long vt_bs, int vt_ld,
                                  float* __restrict__ O, long o_bs, int o_ld,
                                  int nk, float scale,
                                  const float* __restrict__ interf) {
  __shared__ float pbuf_all[4][16 * 16];
  int lane = threadIdx.x & 31;
  int half = lane >> 4, l15 = lane & 15;
  int wid = threadIdx.x >> 5;
  float* pbuf = pbuf_all[wid];
  int tq = blockIdx.x * 4 + wid;
  int h = blockIdx.y, b = blockIdx.z;
  const float* Qb = Q + (size_t)b * q_bs + h * 64;
  const float* Kb = Kp + (size_t)b * k_bs + h * 64;
  const float* VTb = VT + (size_t)b * vt_bs + (size_t)(h * 64) * vt_ld;
  float* Ob = O + (size_t)b * o_bs + h * 64;

  // preload Q tile (pre-scaled) in A-fragment layout: 16 K-chunks of 4
  v2f qa[16];
#pragma unroll
  for (int kc = 0; kc < 16; kc++) {
    v2f q = *(const v2f*)(Qb + (size_t)(tq * 16 + l15) * q_ld + kc * 4 + 2 * half);
    q[0] *= scale;
    q[1] *= scale;
    qa[kc] = q;
  }
  v8f o0 = {0, 0, 0, 0, 0, 0, 0, 0};
  v8f o1 = o0, o2 = o0, o3 = o0;
  float mrow[8], lrow[8];
#pragma unroll
  for (int r = 0; r < 8; r++) { mrow[r] = -1e30f; lrow[r] = 0.f; }

  const float* vt0 = VTb + (size_t)l15 * vt_ld;
  size_t vstep = (size_t)16 * vt_ld;
  int ntiles = (nk + 15) >> 4;
  for (int tj = 0; tj < ntiles; tj++) {
    int jbase = tj * 16;
    // S = (Q*scale) @ K^T : B-frag of K^T is contiguous per lane.
    // Software-pipelined: next K-fragment load in flight during current WMMA.
    v8f s = {0, 0, 0, 0, 0, 0, 0, 0};
    int krow = jbase + l15;
    int krow_c = krow < nk ? krow : (nk - 1);
    const float* kr = Kb + (size_t)krow_c * k_ld + 2 * half;
    v2f kb_c = *(const v2f*)(kr);
#pragma unroll
    for (int kc = 1; kc < 16; kc++) {
      v2f kb_n = *(const v2f*)(kr + kc * 4);
      s = wmma4(qa[kc - 1], kb_c, s);
      kb_c = kb_n;
    }
    s = wmma4(qa[15], kb_c, s);
    // add interference bias / mask out-of-range keys
    int j = jbase + l15;
    bool jval = j < nk;
#pragma unroll
    for (int r = 0; r < 8; r++) {
      if (interf) {
        int i = tq * 16 + r + 8 * half;
        s[r] += interf[(size_t)i * NSPAT + j];
      }
      if (!jval) s[r] = -1e30f;
    }
    // online softmax (row reductions across the 16 lanes of each half)
    float alpha[8];
#pragma unroll
    for (int r = 0; r < 8; r++) {
      float mx = redmax16(s[r]);
      float mn = fmaxf(mrow[r], mx);
      float al = expf(mrow[r] - mn);
      float p = expf(s[r] - mn);
      s[r] = p;
      float rs = redsum16(p);
      lrow[r] = lrow[r] * al + rs;
      mrow[r] = mn;
      alpha[r] = al;
    }
#pragma unroll
    for (int r = 0; r < 8; r++) {
      o0[r] *= alpha[r];
      o1[r] *= alpha[r];
      o2[r] *= alpha[r];
      o3[r] *= alpha[r];
    }
    // re-layout P (C layout) -> A fragments via this wave's LDS slab
#pragma unroll
    for (int r = 0; r < 8; r++) pbuf[(r + 8 * half) * 16 + l15] = s[r];
    v2f pa[4];
#pragma unroll
    for (int kc = 0; kc < 4; kc++)
      pa[kc] = *(const v2f*)(pbuf + l15 * 16 + kc * 4 + 2 * half);
    // O += P @ V, V^T B-fragments contiguous per lane; double-buffered.
    {
      const float* vk = vt0 + jbase + 2 * half;
      v2f v0_c = *(const v2f*)(vk);
      v2f v1_c = *(const v2f*)(vk + vstep);
      v2f v2_c = *(const v2f*)(vk + 2 * vstep);
      v2f v3_c = *(const v2f*)(vk + 3 * vstep);
#pragma unroll
      for (int kc = 1; kc < 4; kc++) {
        v2f v0_n = *(const v2f*)(vk + kc * 4);
        v2f v1_n = *(const v2f*)(vk + vstep + kc * 4);
        v2f v2_n = *(const v2f*)(vk + 2 * vstep + kc * 4);
        v2f v3_n = *(const v2f*)(vk + 3 * vstep + kc * 4);
        o0 = wmma4(pa[kc - 1], v0_c, o0);
        o1 = wmma4(pa[kc - 1], v1_c, o1);
        o2 = wmma4(pa[kc - 1], v2_c, o2);
        o3 = wmma4(pa[kc - 1], v3_c, o3);
        v0_c = v0_n;
        v1_c = v1_n;
        v2_c = v2_n;
        v3_c = v3_n;
      }
      o0 = wmma4(pa[3], v0_c, o0);
      o1 = wmma4(pa[3], v1_c, o1);
      o2 = wmma4(pa[3], v2_c, o2);
      o3 = wmma4(pa[3], v3_c, o3);
    }
  }
  // normalize and store
#pragma unroll
  for (int r = 0; r < 8; r++) {
    float inv = 1.0f / lrow[r];
    int m = tq * 16 + r + 8 * half;
    float* orow = Ob + (size_t)m * o_ld + l15;
    orow[0]  = o0[r] * inv;
    orow[16] = o1[r] * inv;
    orow[32] = o2[r] * inv;
    orow[48] = o3[r] * inv;
  }
}

// ---------------------------------------------------------------------------
extern "C" void kernel_launch(void* const* d_in, const int* in_sizes, int n_in,
                              void* d_out, int out_size, void* d_ws,
                              size_t ws_size, hipStream_t stream) {
  const float* x        = (const float*)d_in[0];   // [4,512,32,32]
  const float* context  = (const float*)d_in[1];   // [4,77,768]
  const float* gn1_w    = (const float*)d_in[2];
  const float* gn1_b    = (const float*)d_in[3];
  const float* gn2_w    = (const float*)d_in[4];
  const float* gn2_b    = (const float*)d_in[5];
  const float* fa_qkv_w = (const float*)d_in[6];   // [512,1536]
  const float* fa_out_w = (const float*)d_in[7];   // [512,512]
  const float* fa_out_b = (const float*)d_in[8];
  const float* wavelen  = (const float*)d_in[9];   // scalar
  const float* ca_q_w   = (const float*)d_in[10];  // [512,512]
  const float* ca_k_w   = (const float*)d_in[11];  // [768,512]
  const float* ca_v_w   = (const float*)d_in[12];  // [768,512]
  const float* ca_out_w = (const float*)d_in[13];  // [512,512]
  const float* ca_out_b = (const float*)d_in[14];
  float* out = (float*)d_out;

  const int B = 4;
  float* ws = (float*)d_ws;
  float* A0 = ws;                           // h1t / h2t [B,1024,512]
  float* A1 = A0 + (size_t)2097152;         // qk [B,1024,1024] / attn2 [B,1024,512]
  float* A2 = A1 + (size_t)4194304;         // attn1 / qproj [B,1024,512]
  float* A3 = A2 + (size_t)2097152;         // x2 [B,1024,512]
  float* A4 = A3 + (size_t)2097152;         // kproj [B,77,512]
  float* A5 = A4 + (size_t)157696;          // VT_cross [B,512,80]
  float* A6 = A5 + (size_t)163840;          // interference [1024,1024]
  float* A7 = A6 + (size_t)1048576;         // VT_fresnel [B,512,1024]
  float* W0 = A7 + (size_t)2097152;         // fa_qkv_w^T [1536,512]
  float* W1 = W0 + (size_t)786432;          // fa_out_w^T [512,512]
  float* W2 = W1 + (size_t)262144;          // ca_q_w^T   [512,512]
  float* W3 = W2 + (size_t)262144;          // ca_k_w^T   [512,768]
  float* W4 = W3 + (size_t)393216;          // ca_v_w^T   [512,768]
  float* W5 = W4 + (size_t)393216;          // ca_out_w^T [512,512]

  // one-time weight transposes (contiguous WMMA B-fragments)
  transpose_kernel<<<dim3(3072), dim3(256), 0, stream>>>(fa_qkv_w, W0, 512, 1536);
  transpose_kernel<<<dim3(1024), dim3(256), 0, stream>>>(fa_out_w, W1, 512, 512);
  transpose_kernel<<<dim3(1024), dim3(256), 0, stream>>>(ca_q_w, W2, 512, 512);
  transpose_kernel<<<dim3(1536), dim3(256), 0, stream>>>(ca_k_w, W3, 768, 512);
  transpose_kernel<<<dim3(1536), dim3(256), 0, stream>>>(ca_v_w, W4, 768, 512);
  transpose_kernel<<<dim3(1024), dim3(256), 0, stream>>>(ca_out_w, W5, 512, 512);
  // interference table (shared by all batches/heads)
  interf_kernel<<<dim3(4096), dim3(256), 0, stream>>>(wavelen, A6);
  // GN1: x [B,C,N] -> A0 [B,N,C]
  groupnorm_kernel<<<dim3(128), dim3(256), 0, stream>>>(x, gn1_w, gn1_b, A0, 0);
  // Q,K projection: [B,1024,512] @ W^T rows 0..1023 -> A1 [B,1024,1024]
  gemm_wmma_kernel<<<dim3(128, 1, B), dim3(256), 0, stream>>>(
      A0, 512, 524288L, W0, 512, nullptr, nullptr, A1, 1024, 1024, 512, 1024, 0);
  // V projection, transposed store -> A7 = V^T [B, 512, 1024]
  gemm_wmma_kernel<<<dim3(64, 1, B), dim3(256), 0, stream>>>(
      A0, 512, 524288L, W0 + (size_t)1024 * 512, 512, nullptr, nullptr, A7, 1024,
      1024, 512, 512, 3);
  // Fresnel self-attention -> A2 [B,1024,512]
  flash_wmma_kernel<<<dim3(16, HEADS, B), dim3(128), 0, stream>>>(
      A1, 1048576L, 1024, A1 + 512, 1048576L, 1024, A7, 524288L, 1024, A2,
      524288L, 512, 1024, 0.125f, A6);
  // out proj + bias + residual(x) -> x2 (A3, [B,N,C])
  gemm_wmma_kernel<<<dim3(64, 1, B), dim3(256), 0, stream>>>(
      A2, 512, 524288L, W1, 512, fa_out_b, x, A3, 512, 1024, 512, 512, 1);
  // GN2: x2 [B,N,C] -> A0 [B,N,C]
  groupnorm_kernel<<<dim3(128), dim3(256), 0, stream>>>(A3, gn2_w, gn2_b, A0, 1);
  // cross-attn projections
  gemm_wmma_kernel<<<dim3(64, 1, B), dim3(256), 0, stream>>>(
      A0, 512, 524288L, W2, 512, nullptr, nullptr, A2, 512, 1024, 512, 512, 0);
  gemm_wmma_kernel<<<dim3(5, 1, B), dim3(256), 0, stream>>>(
      context, 768, 59136L, W3, 768, nullptr, nullptr, A4, 512, 77, 768, 512, 0);
  // V projection, transposed store with padded stride 80 -> A5 [B,512,80]
  gemm_wmma_kernel<<<dim3(5, 1, B), dim3(256), 0, stream>>>(
      context, 768, 59136L, W4, 768, nullptr, nullptr, A5, 80, 77, 768, 512, 3);
  // cross attention -> A1 (reused) [B,1024,512]
  flash_wmma_kernel<<<dim3(16, HEADS, B), dim3(128), 0, stream>>>(
      A2, 524288L, 512, A4, 39424L, 512, A5, 40960L, 80, A1, 524288L, 512, 77,
      0.125f, nullptr);
  // out proj + bias + residual(x2) -> final output [B,C,H,W] (transposed store)
  gemm_wmma_kernel<<<dim3(64, 1, B), dim3(256), 0, stream>>>(
      A1, 512, 524288L, W5, 512, ca_out_b, A3, out, 1024, 1024, 512, 512, 2);
}